// MCR2Variational_3968549781877
// MI455X (gfx1250) — compile-verified
//
#include <hip/hip_runtime.h>
#include <hip/hip_bf16.h>
#include <math.h>

typedef __attribute__((ext_vector_type(2))) float v2f;
typedef __attribute__((ext_vector_type(8))) float v8f;

#define Nn 100000
#define Dd 128
#define Kk 10
#define EPSF 0.5f
#define NSPLIT 50          // blocks per class along N; 100000/50 = 2000 rows each
#define CHUNK 64           // rows of Z staged in LDS per iteration

// CDNA5 async global->LDS copy path (ASYNCcnt). Guarded so host pass and
// older toolchains fall back to the synchronous float4 staging path.
#if __has_builtin(__builtin_amdgcn_global_load_async_to_lds_b128) && \
    __has_builtin(__builtin_amdgcn_global_load_async_to_lds_b32)  && \
    __has_builtin(__builtin_amdgcn_s_wait_asynccnt)
#define HAS_ASYNC 1
#else
#define HAS_ASYNC 0
#endif

#if HAS_ASYNC
// Builtin signatures (from the toolchain diagnostics):
//   b128: (V4i addrspace(1)*, V4i addrspace(3)*, Ii offset, Ii cpol)
//   b32 : (int addrspace(1)*, int addrspace(3)*, Ii offset, Ii cpol)
typedef int async_i4 __attribute__((vector_size(16)));
#define GAS_I4 __attribute__((address_space(1))) async_i4*
#define LAS_I4 __attribute__((address_space(3))) async_i4*
#define GAS_I  __attribute__((address_space(1))) int*
#define LAS_I  __attribute__((address_space(3))) int*
#endif

// ---------------------------------------------------------------------------
// zero workspace
// ---------------------------------------------------------------------------
__global__ void zero_kernel(float* __restrict__ p, int n) {
    int i = blockIdx.x * blockDim.x + threadIdx.x;
    if (i < n) p[i] = 0.f;
}

// ---------------------------------------------------------------------------
// Stage one CHUNK of Z rows + class mask into LDS.
// Full in-range chunks use GLOBAL_LOAD_ASYNC_TO_LDS (no VGPR round trip,
// overlaps with WMMA compute); tail chunk uses guarded sync loads with
// zero-filled mask (mask==0 kills any out-of-slice B-side garbage).
// ---------------------------------------------------------------------------
__device__ __forceinline__ void stage_chunk(const float* __restrict__ Z,
                                            const float* __restrict__ Pi,
                                            int k, int row0, int rows, int c0,
                                            float* zbuf, float* mbuf, int tid) {
    if (c0 >= rows) return;
#if HAS_ASYNC
    if (c0 + CHUNK <= rows) {
        const float4* src = (const float4*)Z + (size_t)(row0 + c0) * 32;
        float4* dst = (float4*)zbuf;
        for (int q = tid; q < CHUNK * 32; q += 256) {
            __builtin_amdgcn_global_load_async_to_lds_b128(
                (GAS_I4)(src + q), (LAS_I4)(dst + q), 0, 0);
        }
        if (tid < CHUNK) {
            __builtin_amdgcn_global_load_async_to_lds_b32(
                (GAS_I)(Pi + (size_t)(row0 + c0 + tid) * Kk + k),
                (LAS_I)(mbuf + tid), 0, 0);
        }
        return;
    }
#endif
    for (int q = tid; q < CHUNK * 32; q += 256) {
        int n = q >> 5, d4 = q & 31;
        float4 val = {0.f, 0.f, 0.f, 0.f};
        if (c0 + n < rows)
            val = ((const float4*)Z)[(size_t)(row0 + c0 + n) * 32 + d4];
        ((float4*)zbuf)[q] = val;
    }
    if (tid < CHUNK) {
        float mv = (c0 + tid < rows) ? Pi[(size_t)(row0 + c0 + tid) * Kk + k] : 0.f;
        mbuf[tid] = mv;
    }
}

__device__ __forceinline__ void wait_stage() {
#if HAS_ASYNC
    __builtin_amdgcn_s_wait_asynccnt(0);
#endif
    __syncthreads();
}

// ---------------------------------------------------------------------------
// Per-class masked Gram: G[k][d][e] = sum_n Pi[n,k] * Z[n,d] * Z[n,e]
// Class-major: Z (51MB) fits in 192MB L2, so K passes over Z are L2-resident.
// Double-buffered async LDS staging overlapped with V_WMMA_F32_16X16X4_F32.
// ---------------------------------------------------------------------------
__global__ __launch_bounds__(256) void gram_kernel(const float* __restrict__ Z,
                                                   const float* __restrict__ Pi,
                                                   float* __restrict__ G,
                                                   float* __restrict__ trPi) {
    __shared__ float Zs[2][CHUNK * Dd];   // 2 x 32 KB double buffer
    __shared__ float ms[2][CHUNK];
    __shared__ float red[256];

    const int blk   = blockIdx.x;
    const int k     = blk / NSPLIT;
    const int split = blk % NSPLIT;
    const int rows  = Nn / NSPLIT;              // 2000
    const int row0  = split * rows;

    const int tid   = threadIdx.x;
    const int lane  = tid & 31;
    const int wave  = tid >> 5;                 // 0..7
    const int dbase = wave * 16;
    const int h     = lane >> 4;                // lane half selects K-pair {0,1} vs {2,3}
    const int l16   = lane & 15;

    v8f acc[8] = {};                            // 8 tiles (16x16 each) per wave
    float msum = 0.f;

    const int nchunks = (rows + CHUNK - 1) / CHUNK;

    stage_chunk(Z, Pi, k, row0, rows, 0, Zs[0], ms[0], tid);
    wait_stage();

    for (int ci = 0; ci < nchunks; ++ci) {
        const int cur = ci & 1;
        // overlap: DMA next chunk into the other buffer while we compute
        stage_chunk(Z, Pi, k, row0, rows, (ci + 1) * CHUNK,
                    Zs[cur ^ 1], ms[cur ^ 1], tid);

        if (tid < CHUNK) msum += ms[cur][tid];

        const float* zb = Zs[cur];
        const float* mb = ms[cur];
        for (int n0 = 0; n0 < CHUNK; n0 += 4) {
            const int na = n0 + 2 * h;
            const float m0 = mb[na], m1 = mb[na + 1];
            const float* za = &zb[na * Dd];
            // A (16x4): A[d][p] = Pi * Z[n,d] ; lanes 0-15 K={0,1}, 16-31 K={2,3}
            v2f a;
            a.x = m0 * za[dbase + l16];
            a.y = m1 * za[Dd + dbase + l16];
#pragma unroll
            for (int ei = 0; ei < 8; ++ei) {
                // B (4x16): B[p][e] = Z[n,e]
                v2f b;
                b.x = za[ei * 16 + l16];
                b.y = za[Dd + ei * 16 + l16];
                acc[ei] = __builtin_amdgcn_wmma_f32_16x16x4_f32(
                    false, a, false, b, (short)0, acc[ei], false, false);
            }
        }
        wait_stage();   // publish chunk ci+1, protect buffer reuse
    }

    // accumulate tiles into global G  (C layout: VGPR v -> row M = v + 8*h, col = l16)
    float* Gk = G + (size_t)k * Dd * Dd;
#pragma unroll
    for (int ei = 0; ei < 8; ++ei)
#pragma unroll
        for (int v = 0; v < 8; ++v) {
            int r = dbase + v + 8 * h;
            int c = ei * 16 + l16;
            atomicAdd(&Gk[r * Dd + c], acc[ei][v]);
        }

    // trPi contribution (mask already staged -> free)
    red[tid] = msum;
    __syncthreads();
    for (int s = 128; s > 0; s >>= 1) {
        if (tid < s) red[tid] += red[tid + s];
        __syncthreads();
    }
    if (tid == 0) atomicAdd(&trPi[k], red[0]);
}

// ---------------------------------------------------------------------------
// Un = U / max(||U[:,p]||, 1e-12)   (column-normalize)
// ---------------------------------------------------------------------------
__global__ __launch_bounds__(128) void normalize_kernel(const float* __restrict__ U,
                                                        float* __restrict__ Un) {
    __shared__ float red[128];
    __shared__ float nrm;
    const int p = blockIdx.x;
    const int d = threadIdx.x;
    float v = U[d * Dd + p];
    red[d] = v * v;
    __syncthreads();
    for (int s = 64; s > 0; s >>= 1) {
        if (d < s) red[d] += red[d + s];
        __syncthreads();
    }
    if (d == 0) nrm = fmaxf(sqrtf(red[0]), 1e-12f);
    __syncthreads();
    Un[d * Dd + p] = v / nrm;
}

// ---------------------------------------------------------------------------
// loss_reg partial: sum_{d,e} (G_k[d][e] - M_k[d][e])^2,
// M_k = (Un * Ar_k) @ Un^T via fp32 WMMA. Un staged in LDS, stride 132
// (=> bank (4d+p)%64, conflict-free for 16-lane column reads).
// ---------------------------------------------------------------------------
__global__ __launch_bounds__(256) void reg_kernel(const float* __restrict__ Un,
                                                  const float* __restrict__ A,
                                                  const float* __restrict__ G,
                                                  float* __restrict__ reg_sum) {
    __shared__ float Uns[Dd * 132];
    __shared__ float Ark[Dd];
    __shared__ float red[256];

    const int k   = blockIdx.x;
    const int tid = threadIdx.x;
    for (int q = tid; q < Dd * Dd; q += 256) {
        int d = q >> 7, p = q & 127;
        Uns[d * 132 + p] = Un[q];
    }
    if (tid < Dd) Ark[tid] = fmaxf(A[k * Dd + tid], 0.f);
    __syncthreads();

    const int lane  = tid & 31;
    const int wave  = tid >> 5;
    const int h     = lane >> 4;
    const int l16   = lane & 15;
    const int dbase = wave * 16;

    v8f acc[8] = {};
    for (int p0 = 0; p0 < Dd; p0 += 4) {
        const int pa = p0 + 2 * h;
        const int dr = dbase + l16;
        v2f a;                                   // A[d][p] = Un[d,p]*Ar[p]
        a.x = Uns[dr * 132 + pa] * Ark[pa];
        a.y = Uns[dr * 132 + pa + 1] * Ark[pa + 1];
#pragma unroll
        for (int ei = 0; ei < 8; ++ei) {
            const int e = ei * 16 + l16;
            v2f b;                               // B[p][e] = Un[e,p]
            b.x = Uns[e * 132 + pa];
            b.y = Uns[e * 132 + pa + 1];
            acc[ei] = __builtin_amdgcn_wmma_f32_16x16x4_f32(
                false, a, false, b, (short)0, acc[ei], false, false);
        }
    }

    const float* Gk = G + (size_t)k * Dd * Dd;
    float local = 0.f;
#pragma unroll
    for (int ei = 0; ei < 8; ++ei)
#pragma unroll
        for (int v = 0; v < 8; ++v) {
            int r = dbase + v + 8 * h;
            int c = ei * 16 + l16;
            float diff = Gk[r * Dd + c] - acc[ei][v];
            local += diff * diff;
        }
    red[tid] = local;
    __syncthreads();
    for (int s = 128; s > 0; s >>= 1) {
        if (tid < s) red[tid] += red[tid + s];
        __syncthreads();
    }
    if (tid == 0) atomicAdd(reg_sum, red[0]);
}

// ---------------------------------------------------------------------------
// finalize: S = sum_k G_k ; Cholesky logdet of I + c*S ; loss_Rc ; outputs.
// ---------------------------------------------------------------------------
__global__ __launch_bounds__(256) void finalize_kernel(const float* __restrict__ G,
                                                       const float* __restrict__ trPi,
                                                       const float* __restrict__ Ain,
                                                       const float* __restrict__ reg_sum,
                                                       float* __restrict__ out) {
    __shared__ float Bm[Dd * 129];   // stride 129 -> bank = (i + j) % 64
    __shared__ float red[256];
    __shared__ float piv;
    __shared__ float lossR_s;

    const int tid = threadIdx.x;
    const float cfac = (float)Dd / ((float)Nn * EPSF);

    for (int q = tid; q < Dd * Dd; q += 256) {
        int d = q >> 7, e = q & 127;
        float s = 0.f;
        for (int k = 0; k < Kk; ++k) s += G[(size_t)k * Dd * Dd + q];
        Bm[d * 129 + e] = cfac * s + ((d == e) ? 1.0f : 0.0f);
    }
    __syncthreads();

    // right-looking Cholesky (lower triangle)
    for (int j = 0; j < Dd; ++j) {
        if (tid == 0) {
            float v = sqrtf(Bm[j * 129 + j]);
            piv = v;
            Bm[j * 129 + j] = v;
        }
        __syncthreads();
        for (int i = j + 1 + tid; i < Dd; i += 256) Bm[i * 129 + j] /= piv;
        __syncthreads();
        for (int i = j + 1 + tid; i < Dd; i += 256) {
            float lij = Bm[i * 129 + j];
            for (int l = j + 1; l <= i; ++l)
                Bm[i * 129 + l] -= lij * Bm[l * 129 + j];
        }
        __syncthreads();
    }

    float part = 0.f;
    for (int j = tid; j < Dd; j += 256) part += logf(Bm[j * 129 + j]);
    red[tid] = part;
    __syncthreads();
    for (int s = 128; s > 0; s >>= 1) {
        if (tid < s) red[tid] += red[tid + s];
        __syncthreads();
    }
    if (tid == 0) lossR_s = red[0];   // 0.5 * 2 * sum(log diag)
    __syncthreads();

    // loss_Rc
    float rc = 0.f;
    if (tid < Kk) {
        float tp = trPi[tid];
        float sc = (float)Dd / (tp * EPSF);
        float s = 0.f;
        for (int p = 0; p < Dd; ++p) {
            float ar = fmaxf(Ain[tid * Dd + p], 0.f);
            s += log1pf(sc * ar);
        }
        rc = tp * s;
    }
    red[tid] = rc;
    __syncthreads();
    for (int s = 128; s > 0; s >>= 1) {
        if (tid < s) red[tid] += red[tid + s];
        __syncthreads();
    }
    if (tid == 0) {
        float loss_R   = lossR_s;
        float loss_Rc  = 0.5f * red[0] / (float)Nn;
        float loss_reg = 0.5f * reg_sum[0];
        float loss_obj = loss_R - loss_Rc - 1.0f * loss_reg;   // MU = 1
        out[0] = -loss_obj;
        out[1] = loss_R;
        out[2] = loss_Rc;
        out[3] = loss_reg;
    }
}

// ---------------------------------------------------------------------------
extern "C" void kernel_launch(void* const* d_in, const int* in_sizes, int n_in,
                              void* d_out, int out_size, void* d_ws, size_t ws_size,
                              hipStream_t stream) {
    const float* Z  = (const float*)d_in[0];   // [N, D]
    const float* Pi = (const float*)d_in[1];   // [N, K]
    const float* A  = (const float*)d_in[2];   // [K, D]
    const float* U  = (const float*)d_in[3];   // [D, D]
    float* out = (float*)d_out;                // 4 scalars
    float* ws  = (float*)d_ws;

    // workspace layout (floats)
    float* G       = ws;                 // 10*128*128 = 163840
    float* trPi    = ws + 163840;        // 16
    float* reg_sum = ws + 163856;        // 16 (use [0])
    float* Un      = ws + 163872;        // 16384

    const int nzero = 163872;            // G + trPi + reg_sum
    zero_kernel<<<(nzero + 255) / 256, 256, 0, stream>>>(ws, nzero);
    gram_kernel<<<Kk * NSPLIT, 256, 0, stream>>>(Z, Pi, G, trPi);
    normalize_kernel<<<Dd, 128, 0, stream>>>(U, Un);
    reg_kernel<<<Kk, 256, 0, stream>>>(Un, A, G, reg_sum);
    finalize_kernel<<<1, 256, 0, stream>>>(G, trPi, A, reg_sum, out);
}